// SparseBatchAutoEncoder_73899207295094
// MI455X (gfx1250) — compile-verified
//
#include <hip/hip_runtime.h>
#include <stdint.h>

// ---------------------------------------------------------------------------
// SparseBatchAutoEncoder forward, fp32 end-to-end via V_WMMA_F32_16X16X4_F32.
// B=512, K=20000, EMB=256, HID=512, NUM_EMB=200000.
// ---------------------------------------------------------------------------

typedef float v2f __attribute__((ext_vector_type(2)));
typedef float v8f __attribute__((ext_vector_type(8)));
typedef int b128_t __attribute__((vector_size(16)));  // matches builtin proto

#define AS1 __attribute__((address_space(1)))
#define AS3 __attribute__((address_space(3)))

#if defined(__HIP_DEVICE_COMPILE__)
#  if !__has_builtin(__builtin_amdgcn_wmma_f32_16x16x4_f32)
#    error "device pass: __builtin_amdgcn_wmma_f32_16x16x4_f32 not declared"
#  endif
#  define WMMA_F32X4(a, b, c) \
     __builtin_amdgcn_wmma_f32_16x16x4_f32(false, (a), false, (b), (short)0, (c), false, false)
#  if __has_builtin(__builtin_amdgcn_global_load_async_to_lds_b128)
#    define HAVE_ASYNC_LDS 1
#  else
#    define HAVE_ASYNC_LDS 0
#  endif
#else
#  define WMMA_F32X4(a, b, c) (c)  // host pass: never executed
#  define HAVE_ASYNC_LDS 0
#endif

// selu(x) = scale * (x > 0 ? x : alpha*(exp(x)-1))
__device__ __forceinline__ float selu_f(float x) {
  const float sc = 1.0507009873554805f;
  const float sa = 1.7580993408473768f;  // sc * alpha
  return x > 0.0f ? sc * x : sa * (__expf(x) - 1.0f);
}

// ---------------------------------------------------------------------------
// K1: z1 = selu(RI[512,20000] @ en_emb[active_dims][20000,256] + en_bias)
// One 16x16 output tile per block; 8 waves split K (2500 each); LDS reduce.
// ---------------------------------------------------------------------------
__global__ __launch_bounds__(256) void k1_sparse_encode(
    const float* __restrict__ RI,       // [512, 20000]
    const int* __restrict__ idx,        // [20000]
    const float* __restrict__ en_emb,   // [200000, 256]
    const float* __restrict__ en_bias,  // [256]
    float* __restrict__ z1)             // [512, 256]
{
  __shared__ float red[8 * 256];
  const int lane = threadIdx.x & 31;
  const int wave = threadIdx.x >> 5;
  const int ml = lane & 15;   // M (A) / N (B) index within half-wave
  const int kh = lane >> 4;   // half-wave select: K offset {0,2}
  const int n0 = blockIdx.x * 16;  // EMB tile
  const int m0 = blockIdx.y * 16;  // batch tile

  v8f c = {};
  const float* arow = RI + (size_t)(m0 + ml) * 20000 + 2 * kh;
  const uint32_t col = (uint32_t)(n0 + ml);
  const int kbeg = wave * 2500;
  for (int kk = kbeg; kk < kbeg + 2500; kk += 4) {
    v2f a = *(const v2f*)(arow + kk);            // RI[m0+ml][kk+2kh .. +1]
    const int4 iv = *(const int4*)(idx + kk);    // one 16B broadcast request
    const int r0 = kh ? iv.z : iv.x;
    const int r1 = kh ? iv.w : iv.y;
    v2f b;                                       // B[k][n] = en_emb[idx[k]][n]
    b.x = en_emb[(uint32_t)r0 * 256u + col];     // 32-bit voffset + sgpr base
    b.y = en_emb[(uint32_t)r1 * 256u + col];
    c = WMMA_F32X4(a, b, c);
  }

  // C layout: VGPR j -> (m = j + 8*kh, n = ml)
  #pragma unroll
  for (int j = 0; j < 8; ++j) {
    const int m = j + 8 * kh;
    red[wave * 256 + m * 16 + ml] = c[j];
  }
  __syncthreads();

  const int t = threadIdx.x;         // element (m = t/16, n = t%16)
  float s = 0.0f;
  #pragma unroll
  for (int w = 0; w < 8; ++w) s += red[w * 256 + t];
  const int m = t >> 4, n = t & 15;
  s += en_bias[n0 + n];
  z1[(size_t)(m0 + m) * 256 + n0 + n] = selu_f(s);
}

// ---------------------------------------------------------------------------
// K2/K3: out[512,Nd] = selu(A[512,Kd] @ W[Nd,Kd]^T + bias). One wave per tile.
// nshift = log2(Nd/16).
// ---------------------------------------------------------------------------
__global__ __launch_bounds__(256) void k_dense_selu(
    const float* __restrict__ A,      // [512, Kd]
    const float* __restrict__ W,      // [Nd, Kd] (row-major, used transposed)
    const float* __restrict__ bias,   // [Nd]
    float* __restrict__ out,          // [512, Nd]
    int Nd, int Kd, int nshift)
{
  const int lane = threadIdx.x & 31;
  const int wave = threadIdx.x >> 5;
  const int ml = lane & 15;
  const int kh = lane >> 4;
  const int tile = blockIdx.x * 8 + wave;
  const int mt = tile >> nshift;
  const int nt = tile & ((1 << nshift) - 1);
  const int m0 = mt * 16, n0 = nt * 16;

  v8f c = {};
  const float* arow = A + (size_t)(m0 + ml) * Kd + 2 * kh;
  const float* wrow = W + (size_t)(n0 + ml) * Kd + 2 * kh;  // B[k][n] = W[n][k]
  for (int kk = 0; kk < Kd; kk += 4) {
    v2f a = *(const v2f*)(arow + kk);
    v2f b = *(const v2f*)(wrow + kk);
    c = WMMA_F32X4(a, b, c);
  }

  const float bn = bias[n0 + ml];
  #pragma unroll
  for (int j = 0; j < 8; ++j) {
    const int m = m0 + j + 8 * kh;
    out[(size_t)m * Nd + n0 + ml] = selu_f(c[j] + bn);
  }
}

// ---------------------------------------------------------------------------
// K4: out[512,200000] = z3[512,256] @ de_emb[200000,256]^T + de_bias (no act).
// One 16-wide N tile per block; de_emb slab async-staged to LDS once (padded
// rows); 8 waves x 4 accumulators sweep all 32 M tiles, sharing B fragments.
// ---------------------------------------------------------------------------
#define LDS_STRIDE 260  // 256 + 4 pad -> fragment reads hit 32 distinct banks

__global__ __launch_bounds__(256) void k_decode_full(
    const float* __restrict__ z3,       // [512, 256]
    const float* __restrict__ de_emb,   // [200000, 256]
    const float* __restrict__ de_bias,  // [200000]
    float* __restrict__ out)            // [512, 200000]
{
  __shared__ float tile[16 * LDS_STRIDE];
  const int n0 = blockIdx.x * 16;

  // cooperative stage: 16 rows x 256 cols of de_emb -> LDS, 16B chunks.
  // On gfx1250 this uses GLOBAL_LOAD_ASYNC_TO_LDS_B128 (no VGPR round-trip,
  // tracked on ASYNCcnt); falls back to load+ds_store if builtin is absent.
  for (int c4 = threadIdx.x; c4 < 16 * 64; c4 += 256) {
    const int row = c4 >> 6;
    const int col = (c4 & 63) << 2;
    const float* g = de_emb + (size_t)(n0 + row) * 256 + col;
    float* l = &tile[row * LDS_STRIDE + col];
#if HAVE_ASYNC_LDS
    // amdgcn: low 32 bits of a generic LDS pointer are the LDS offset.
    __builtin_amdgcn_global_load_async_to_lds_b128(
        (AS1 b128_t*)(uintptr_t)g, (AS3 b128_t*)(uintptr_t)l, 0, 0);
#else
    *(float4*)l = *(const float4*)g;
#endif
  }
#if HAVE_ASYNC_LDS
#  if __has_builtin(__builtin_amdgcn_s_wait_asynccnt)
  __builtin_amdgcn_s_wait_asynccnt(0);
#  else
  asm volatile("s_wait_asynccnt 0x0" ::: "memory");
#  endif
#endif
  __syncthreads();

  const int lane = threadIdx.x & 31;
  const int wave = threadIdx.x >> 5;
  const int ml = lane & 15;
  const int kh = lane >> 4;

  v8f c0 = {}, c1 = {}, c2 = {}, c3 = {};
  // m-tiles handled by this wave: wave, wave+8, wave+16, wave+24
  const float* a0 = z3 + (size_t)((wave + 0) * 16 + ml) * 256 + 2 * kh;
  const float* a1 = z3 + (size_t)((wave + 8) * 16 + ml) * 256 + 2 * kh;
  const float* a2 = z3 + (size_t)((wave + 16) * 16 + ml) * 256 + 2 * kh;
  const float* a3 = z3 + (size_t)((wave + 24) * 16 + ml) * 256 + 2 * kh;
  const float* tl = tile + ml * LDS_STRIDE + 2 * kh;  // B[k][n] = de_emb[n][k]

  for (int kk = 0; kk < 256; kk += 4) {
    const v2f b  = *(const v2f*)(tl + kk);
    const v2f f0 = *(const v2f*)(a0 + kk);
    const v2f f1 = *(const v2f*)(a1 + kk);
    const v2f f2 = *(const v2f*)(a2 + kk);
    const v2f f3 = *(const v2f*)(a3 + kk);
    c0 = WMMA_F32X4(f0, b, c0);
    c1 = WMMA_F32X4(f1, b, c1);
    c2 = WMMA_F32X4(f2, b, c2);
    c3 = WMMA_F32X4(f3, b, c3);
  }

  const float bn = de_bias[n0 + ml];
  #pragma unroll
  for (int j = 0; j < 8; ++j) {
    const int mr = j + 8 * kh;
    out[(size_t)((wave + 0) * 16 + mr) * 200000 + n0 + ml] = c0[j] + bn;
    out[(size_t)((wave + 8) * 16 + mr) * 200000 + n0 + ml] = c1[j] + bn;
    out[(size_t)((wave + 16) * 16 + mr) * 200000 + n0 + ml] = c2[j] + bn;
    out[(size_t)((wave + 24) * 16 + mr) * 200000 + n0 + ml] = c3[j] + bn;
  }
}

// ---------------------------------------------------------------------------
extern "C" void kernel_launch(void* const* d_in, const int* in_sizes, int n_in,
                              void* d_out, int out_size, void* d_ws, size_t ws_size,
                              hipStream_t stream) {
  const float* RI      = (const float*)d_in[0];  // [512, 20000]
  const int*   idx     = (const int*)  d_in[1];  // [20000]
  const float* en_emb  = (const float*)d_in[2];  // [200000, 256]
  const float* en_bias = (const float*)d_in[3];  // [256]
  const float* enc_W   = (const float*)d_in[4];  // [512, 256]
  const float* enc_b   = (const float*)d_in[5];  // [512]
  const float* dec_W   = (const float*)d_in[6];  // [256, 512]
  const float* dec_b   = (const float*)d_in[7];  // [256]
  const float* de_emb  = (const float*)d_in[8];  // [200000, 256]
  const float* de_bias = (const float*)d_in[9];  // [200000]
  float* out = (float*)d_out;                    // [512, 200000]

  float* z1 = (float*)d_ws;        // [512, 256]
  float* z2 = z1 + 512 * 256;      // [512, 512]
  float* z3 = z2 + 512 * 512;      // [512, 256]

  hipLaunchKernelGGL(k1_sparse_encode, dim3(16, 32), dim3(256), 0, stream,
                     RI, idx, en_emb, en_bias, z1);
  hipLaunchKernelGGL(k_dense_selu, dim3(128), dim3(256), 0, stream,
                     z1, enc_W, enc_b, z2, 512, 256, 5);
  hipLaunchKernelGGL(k_dense_selu, dim3(64), dim3(256), 0, stream,
                     z2, dec_W, dec_b, z3, 256, 512, 4);
  hipLaunchKernelGGL(k_decode_full, dim3(12500), dim3(256), 0, stream,
                     z3, de_emb, de_bias, out);
}